// LIF_22016002360057
// MI455X (gfx1250) — compile-verified
//
#include <hip/hip_runtime.h>
#include <stdint.h>

// ---------------------------------------------------------------------------
// LIF scan over T=16 (innermost, contiguous). Memory-bound: 537MB => ~23us
// floor at 23.3 TB/s. Strategy: per-wave async global->LDS staging with a
// padded transpose so global traffic is perfectly coalesced (512B dense per
// async b128 instruction) and LDS readback is 16B-aligned with only 2-way
// bank conflicts (80B = 20-dword neuron stride).
// ---------------------------------------------------------------------------

#define LIF_DECAY  0.5f
#define LIF_THRESH 1.0f

#define T_STEPS          16
#define NEURON_BYTES     64            // 16 * fp32
#define NEURON_STRIDE_B  80            // padded LDS stride (20 dwords)
#define WAVE_NEURONS     32
#define WAVE_LDS_B       (WAVE_NEURONS * NEURON_STRIDE_B)  // 2560 B per wave
#define WAVES_PER_BLOCK  8
#define BLOCK_THREADS    256

#if defined(__AMDGCN__) && \
    __has_builtin(__builtin_amdgcn_global_load_async_to_lds_b128) && \
    __has_builtin(__builtin_amdgcn_global_store_async_from_lds_b128)
#define USE_ASYNC_LDS 1
#else
#define USE_ASYNC_LDS 0
#endif

typedef int   v4i __attribute__((vector_size(16)));
typedef float v4f __attribute__((vector_size(16)));

typedef __attribute__((address_space(1))) char glob_char;
typedef __attribute__((address_space(3))) char lds_char;
typedef __attribute__((address_space(1))) v4i  glob_v4i;
typedef __attribute__((address_space(3))) v4i  lds_v4i;

__device__ __forceinline__ void wait_asynccnt0() {
#if defined(__AMDGCN__)
#if __has_builtin(__builtin_amdgcn_s_wait_asynccnt)
  __builtin_amdgcn_s_wait_asynccnt(0);
#else
  asm volatile("s_wait_asynccnt 0x0" ::: "memory");
#endif
#endif
}

__device__ __forceinline__ void wait_dscnt0() {
#if defined(__AMDGCN__)
#if __has_builtin(__builtin_amdgcn_s_wait_dscnt)
  __builtin_amdgcn_s_wait_dscnt(0);
#else
  asm volatile("s_wait_dscnt 0x0" ::: "memory");
#endif
#endif
}

__global__ __launch_bounds__(BLOCK_THREADS)
void lif_22016002360057_kernel(const float* __restrict__ X,
                               float* __restrict__ OUT,
                               long long n_neurons) {
  __shared__ __align__(16) char smem[WAVES_PER_BLOCK * WAVE_LDS_B];

  const int wave = threadIdx.x >> 5;
  const int lane = threadIdx.x & 31;

  const long long waveNeuron0 =
      ((long long)blockIdx.x * WAVES_PER_BLOCK + wave) * WAVE_NEURONS;
  if (waveNeuron0 >= n_neurons) return;  // wave-uniform guard

  const size_t waveByteBase = (size_t)waveNeuron0 * NEURON_BYTES;
  char* ldsRegion = smem + wave * WAVE_LDS_B;

  float v[T_STEPS];

#if USE_ASYNC_LDS
  // ---- Stage in: coalesced global -> padded LDS transpose -----------------
  {
    const int n0 = lane >> 2;   // neuron sub-index within 512B span
    const int c  = lane & 3;    // 16B chunk within neuron
    lds_char* lbase = (lds_char*)ldsRegion;
    const glob_char* gbase =
        (glob_char*)((const char*)X + waveByteBase + (size_t)lane * 16);
#pragma unroll
    for (int j = 0; j < 4; ++j) {
      // instruction j: lanes read one dense 512B region of global memory
      __builtin_amdgcn_global_load_async_to_lds_b128(
          (glob_v4i*)(gbase + j * 512),
          (lds_v4i*)(lbase + (size_t)(j * 8 + n0) * NEURON_STRIDE_B + c * 16),
          0, 0);
    }
    wait_asynccnt0();
  }
  // ---- LDS -> registers (16B aligned, 2-way conflicts max) ----------------
  {
    const char* myRow = ldsRegion + (size_t)lane * NEURON_STRIDE_B;
#pragma unroll
    for (int k = 0; k < 4; ++k) {
      v4f t = *(const v4f*)(myRow + k * 16);
      v[4 * k + 0] = t[0]; v[4 * k + 1] = t[1];
      v[4 * k + 2] = t[2]; v[4 * k + 3] = t[3];
    }
  }
#else
  // ---- Fallback: direct strided wide loads --------------------------------
  {
    const char* myRow = (const char*)X + waveByteBase + (size_t)lane * NEURON_BYTES;
#pragma unroll
    for (int k = 0; k < 4; ++k) {
      v4f t = __builtin_nontemporal_load((const v4f*)(myRow + k * 16));
      v[4 * k + 0] = t[0]; v[4 * k + 1] = t[1];
      v[4 * k + 2] = t[2]; v[4 * k + 3] = t[3];
    }
  }
#endif

  // ---- Sequential LIF scan (the only real compute) ------------------------
  float m = 0.0f;
#pragma unroll
  for (int t = 0; t < T_STEPS; ++t) {
    m = m * LIF_DECAY + v[t];
    const bool fire = (m >= LIF_THRESH);   // NaN => no fire, matches reference
    v[t] = fire ? 1.0f : 0.0f;
    m    = fire ? 0.0f : m;
  }

#if USE_ASYNC_LDS
  // ---- registers -> LDS, then coalesced LDS -> global ---------------------
  {
    char* myRow = ldsRegion + (size_t)lane * NEURON_STRIDE_B;
#pragma unroll
    for (int k = 0; k < 4; ++k) {
      v4f t = {v[4 * k + 0], v[4 * k + 1], v[4 * k + 2], v[4 * k + 3]};
      *(v4f*)(myRow + k * 16) = t;
    }
  }
  wait_dscnt0();  // LDS writes visible before the async engine reads them
  {
    const int n0 = lane >> 2;
    const int c  = lane & 3;
    lds_char* lbase = (lds_char*)ldsRegion;
    glob_char* gbase =
        (glob_char*)((char*)OUT + waveByteBase + (size_t)lane * 16);
#pragma unroll
    for (int j = 0; j < 4; ++j) {
      __builtin_amdgcn_global_store_async_from_lds_b128(
          (glob_v4i*)(gbase + j * 512),
          (lds_v4i*)(lbase + (size_t)(j * 8 + n0) * NEURON_STRIDE_B + c * 16),
          0, 0);
    }
    wait_asynccnt0();  // drain before wave end / LDS reuse
  }
#else
  // ---- Fallback: direct strided wide stores -------------------------------
  {
    char* myRow = (char*)OUT + waveByteBase + (size_t)lane * NEURON_BYTES;
#pragma unroll
    for (int k = 0; k < 4; ++k) {
      v4f t = {v[4 * k + 0], v[4 * k + 1], v[4 * k + 2], v[4 * k + 3]};
      __builtin_nontemporal_store(t, (v4f*)(myRow + k * 16));
    }
  }
#endif
}

extern "C" void kernel_launch(void* const* d_in, const int* in_sizes, int n_in,
                              void* d_out, int out_size, void* d_ws, size_t ws_size,
                              hipStream_t stream) {
  (void)n_in; (void)out_size; (void)d_ws; (void)ws_size;
  const float* X  = (const float*)d_in[0];
  float* OUT      = (float*)d_out;

  const long long total    = (long long)in_sizes[0];      // 32*128*32*32*16
  const long long neurons  = total / T_STEPS;             // 4,194,304
  const long long perBlock = (long long)WAVES_PER_BLOCK * WAVE_NEURONS;  // 256
  const int blocks = (int)((neurons + perBlock - 1) / perBlock);         // 16384

  lif_22016002360057_kernel<<<blocks, BLOCK_THREADS, 0, stream>>>(X, OUT, neurons);
}